// DTAModel_v2_14181982011975
// MI455X (gfx1250) — compile-verified
//
#include <hip/hip_runtime.h>
#include <math.h>

#define N_NODES 40960
#define N_EDGES 163840
#define N_GRAPHS 1024
#define SEQ 1000
#define L1 999
#define L2 998
#define L3 997
#define PSTRIDE 1008   // padded L stride (multiple of 16)

typedef __attribute__((ext_vector_type(2))) float v2f;
typedef __attribute__((ext_vector_type(8))) float v8f;

__device__ __forceinline__ v8f wmma4(v2f a, v2f b, v8f c) {
  // D = A(16x4,f32) * B(4x16,f32) + C(16x16,f32)
  return __builtin_amdgcn_wmma_f32_16x16x4_f32(false, a, false, b, (short)0, c,
                                               false, false);
}

__device__ __forceinline__ int iclamp(int v, int lo, int hi) {
  return v < lo ? lo : (v > hi ? hi : v);
}

__device__ __forceinline__ v2f ld2(const float* p) {
  float2 t = *(const float2*)p;
  v2f r; r[0] = t.x; r[1] = t.y;
  return r;
}

// ---------------------------------------------------------------- elementwise
__global__ void k_fill(float* p, float v, int n) {
  int i = blockIdx.x * blockDim.x + threadIdx.x;
  if (i < n) p[i] = v;
}

__global__ void k_deg_count(const int* __restrict__ dst, float* __restrict__ deg, int e) {
  int i = blockIdx.x * blockDim.x + threadIdx.x;
  if (i < e) atomicAdd(&deg[dst[i]], 1.0f);
}

__global__ void k_rsqrt_inplace(float* p, int n) {
  int i = blockIdx.x * blockDim.x + threadIdx.x;
  if (i < n) p[i] = rsqrtf(p[i]);
}

// ------------------------------------------------------------- generic WMMA GEMM
// C[M,N] = A[M,K] @ W[K,N] (+ bias[N]); row-major. N%16==0, M%16==0,
// gridDim.x*64 >= M. B panel staged (zero-padded to Kpad) in LDS.
// Main K loop is unguarded pointer-walking; ragged K handled by one tail step.
__global__ void __launch_bounds__(128) k_gemm(
    const float* __restrict__ A, const float* __restrict__ W,
    const float* __restrict__ bias, float* __restrict__ C,
    int M, int K, int N) {
  __shared__ float s_w[16 * 512];  // [n][kk], Kpad <= 512
  const int Kpad = (K + 3) & ~3;
  const int n0 = blockIdx.y * 16;
  for (int e = threadIdx.x; e < 16 * Kpad; e += 128) {
    int n = e / Kpad, kk = e - n * Kpad;
    int kc = kk < K ? kk : K - 1;
    float v = W[(size_t)kc * N + n0 + n];
    s_w[e] = (kk < K) ? v : 0.f;
  }
  __syncthreads();
  const int w = threadIdx.x >> 5, lane = threadIdx.x & 31;
  const int lg = lane >> 4, ml = lane & 15;
  const int m0 = (blockIdx.x * 4 + w) * 16;
  if (m0 >= M) return;
  const float* __restrict__ arow = A + (size_t)(m0 + ml) * K;
  const float* __restrict__ brow = s_w + ml * Kpad;

  v8f acc = {};
  const int nfull = K >> 2;          // unguarded 4-deep steps
  const float* ap = arow + 2 * lg;
  const float* bp = brow + 2 * lg;
  if (nfull > 0) {
    v2f a, b;
    a[0] = ap[0]; a[1] = ap[1];
    b = ld2(bp);
    for (int it = 1; it < nfull; ++it) {
      ap += 4; bp += 4;
      v2f an, bn;
      an[0] = ap[0]; an[1] = ap[1];
      bn = ld2(bp);
      acc = wmma4(a, b, acc);
      a = an; b = bn;
    }
    acc = wmma4(a, b, acc);
  }
  if (K & 3) {                       // single guarded tail (only the K=9 GEMM)
    int k0 = nfull << 2;
    v2f a, b;
#pragma unroll
    for (int v = 0; v < 2; ++v) {
      int kk = k0 + 2 * lg + v;
      int kc = kk < K ? kk : K - 1;
      float av = arow[kc];
      a[v] = (kk < K) ? av : 0.f;
    }
    b = ld2(&brow[k0 + 2 * lg]);     // LDS panel is zero-padded
    acc = wmma4(a, b, acc);
  }

#pragma unroll
  for (int r = 0; r < 8; ++r) {
    int m = m0 + r + 8 * lg;
    float o = acc[r];
    if (bias) o += bias[n0 + ml];
    C[(size_t)m * N + n0 + ml] = o;
  }
}

// ------------------------------------------------------------------ GCN pieces
__global__ void k_agg_init(const float* __restrict__ h, const float* __restrict__ dinv,
                           const float* __restrict__ bias, float* __restrict__ out,
                           int fshift) {
  int i = blockIdx.x * blockDim.x + threadIdx.x;
  if (i >= (N_NODES << fshift)) return;
  int n = i >> fshift, f = i & ((1 << fshift) - 1);
  float d = dinv[n];
  out[i] = bias[f] + h[i] * d * d;
}

__global__ void k_agg_edge(const float* __restrict__ h, const float* __restrict__ dinv,
                           const int* __restrict__ src, const int* __restrict__ dst,
                           float* __restrict__ out, int fshift) {
  int i = blockIdx.x * blockDim.x + threadIdx.x;
  if (i >= (N_EDGES << fshift)) return;
  int e = i >> fshift, f = i & ((1 << fshift) - 1);
  int s = src[e], d = dst[e];
  atomicAdd(&out[(d << fshift) + f], h[(s << fshift) + f] * dinv[s] * dinv[d]);
}

// training-mode BN stats: one block per feature, f64 accumulation
__global__ void __launch_bounds__(256) k_bn_stats(
    const float* __restrict__ X, float* __restrict__ mean, float* __restrict__ rstd,
    int M, int F) {
  __shared__ double ls[256], lq[256];
  int f = blockIdx.x;
  double s = 0.0, sq = 0.0;
  for (int r = threadIdx.x; r < M; r += blockDim.x) {
    double v = (double)X[(size_t)r * F + f];
    s += v; sq += v * v;
  }
  ls[threadIdx.x] = s; lq[threadIdx.x] = sq;
  __syncthreads();
  for (int off = 128; off > 0; off >>= 1) {
    if (threadIdx.x < off) {
      ls[threadIdx.x] += ls[threadIdx.x + off];
      lq[threadIdx.x] += lq[threadIdx.x + off];
    }
    __syncthreads();
  }
  if (threadIdx.x == 0) {
    double m = ls[0] / M;
    double var = lq[0] / M - m * m;
    mean[f] = (float)m;
    rstd[f] = rsqrtf((float)var + 1e-5f);
  }
}

__global__ void k_bn_relu(float* __restrict__ X, const float* __restrict__ g,
                          const float* __restrict__ be, const float* __restrict__ mean,
                          const float* __restrict__ rstd, int total, int fmask) {
  int i = blockIdx.x * blockDim.x + threadIdx.x;
  if (i >= total) return;
  int f = i & fmask;
  float v = (X[i] - mean[f]) * rstd[f] * g[f] + be[f];
  X[i] = fmaxf(v, 0.f);
}

// global mean pool into z[:, 0:128]
__global__ void k_pool(const float* __restrict__ act, const int* __restrict__ batch,
                       float* __restrict__ z, float* __restrict__ cnt) {
  int i = blockIdx.x * blockDim.x + threadIdx.x;
  if (i >= N_NODES * 128) return;
  int n = i >> 7, f = i & 127;
  int b = batch[n];
  atomicAdd(&z[b * 224 + f], act[i]);
  if (f == 0) atomicAdd(&cnt[b], 1.0f);
}

__global__ void k_pool_div(float* __restrict__ z, const float* __restrict__ cnt) {
  int i = blockIdx.x * blockDim.x + threadIdx.x;
  if (i >= N_GRAPHS * 128) return;
  int b = i >> 7, f = i & 127;
  z[b * 224 + f] /= fmaxf(cnt[b], 1.0f);
}

// --------------------------------------------------- protein conv1 (fused embed)
// kk = kp*128 + ci. Outer kp loop hoists pos / ps lookup / row pointers; the
// inner ci loop is two float2 LDS loads + wmma.
__global__ void __launch_bounds__(128) k_conv1(
    const int* __restrict__ ps, const float* __restrict__ emb,
    const float* __restrict__ cw, const float* __restrict__ cb,
    float* __restrict__ p1) {
  __shared__ float s_emb[26 * 128];
  __shared__ float s_w[16 * 512];
  __shared__ int s_ps[SEQ];
  const int b = blockIdx.z, co0 = blockIdx.y * 16, tid = threadIdx.x;
  for (int e = tid; e < 26 * 128; e += 128) s_emb[e] = emb[e];
  for (int t = tid; t < SEQ; t += 128) s_ps[t] = ps[b * SEQ + t];
  for (int e = tid; e < 16 * 512; e += 128) {
    int co = e >> 9, kk = e & 511;
    int kp = kk >> 7, ci = kk & 127;
    s_w[e] = cw[(co0 + co) * 512 + ci * 4 + kp];
  }
  __syncthreads();
  const int w = tid >> 5, lane = tid & 31;
  const int lg = lane >> 4, ml = lane & 15;
  const int lt = blockIdx.x * 4 + w;
  if (lt >= 63) return;
  const int l0 = lt * 16;
  v8f acc = {};
#pragma unroll
  for (int kp = 0; kp < 4; ++kp) {
    int pos = l0 + ml + kp - 1;
    int pc = iclamp(pos, 0, SEQ - 1);
    bool ok = (pos >= 0) && (pos < SEQ);
    const float* __restrict__ er = s_emb + s_ps[pc] * 128 + 2 * lg;
    const float* __restrict__ br = s_w + ml * 512 + kp * 128 + 2 * lg;
#pragma unroll 8
    for (int c = 0; c < 128; c += 4) {
      v2f av = ld2(&er[c]);
      v2f a; a[0] = ok ? av[0] : 0.f; a[1] = ok ? av[1] : 0.f;
      v2f bb = ld2(&br[c]);
      acc = wmma4(a, bb, acc);
    }
  }
#pragma unroll
  for (int r = 0; r < 8; ++r) {
    int l = l0 + r + 8 * lg;
    if (l < L1) {
      int co = co0 + ml;
      p1[(b * 32 + co) * PSTRIDE + l] = fmaxf(acc[r] + cb[co], 0.f);
    }
  }
}

// ------------------------------------------------------------------ conv2
// kk = kp*32 + ci. Input halo tile staged [t][ci]; outer kp, inner ci.
__global__ void __launch_bounds__(128) k_conv2(
    const float* __restrict__ p1, const float* __restrict__ cw,
    const float* __restrict__ cb, float* __restrict__ p2) {
  __shared__ float s_w[16 * 192];
  __shared__ float s_in[72 * 32];  // [t][ci]
  const int b = blockIdx.z, co0 = blockIdx.y * 16, bl0 = blockIdx.x * 64;
  const int tid = threadIdx.x;
  for (int e = tid; e < 16 * 192; e += 128) {
    int co = e / 192, kk = e - co * 192;
    int kp = kk >> 5, ci = kk & 31;
    s_w[e] = cw[(co0 + co) * 192 + ci * 6 + kp];
  }
  for (int e = tid; e < 32 * 72; e += 128) {
    int ci = e / 72, t = e - ci * 72;       // consecutive threads -> consecutive t
    int pos = bl0 - 2 + t;
    int pc = iclamp(pos, 0, L1 - 1);
    float v = p1[(b * 32 + ci) * PSTRIDE + pc];
    s_in[t * 32 + ci] = (pos >= 0 && pos < L1) ? v : 0.f;
  }
  __syncthreads();
  const int w = tid >> 5, lane = tid & 31;
  const int lg = lane >> 4, ml = lane & 15;
  const int l0 = bl0 + w * 16;
  v8f acc = {};
#pragma unroll
  for (int kp = 0; kp < 6; ++kp) {
    const float* __restrict__ ir = s_in + (w * 16 + ml + kp) * 32 + 2 * lg;
    const float* __restrict__ br = s_w + ml * 192 + kp * 32 + 2 * lg;
#pragma unroll
    for (int c = 0; c < 32; c += 4) {
      v2f a = ld2(&ir[c]);
      v2f bb = ld2(&br[c]);
      acc = wmma4(a, bb, acc);
    }
  }
#pragma unroll
  for (int r = 0; r < 8; ++r) {
    int l = l0 + r + 8 * lg;
    if (l < L2) {
      int co = co0 + ml;
      p2[(b * 64 + co) * PSTRIDE + l] = fmaxf(acc[r] + cb[co], 0.f);
    }
  }
}

// -------------------------------------------- conv3 fused with bias+relu+max_l
// kk = kp*64 + ci. Block = (co-tile, b); iterates 64-position L chunks:
// stage halo tile [72][64] in LDS (coalesced), 4 waves compute from LDS.
__global__ void __launch_bounds__(128) k_conv3max(
    const float* __restrict__ p2, const float* __restrict__ cw,
    const float* __restrict__ cb, float* __restrict__ z) {
  __shared__ float s_w[16 * 512];
  __shared__ float s_in[72 * 64];  // [t][ci]
  __shared__ float s_red[4 * 16];
  const int b = blockIdx.y, co0 = blockIdx.x * 16, tid = threadIdx.x;
  for (int e = tid; e < 16 * 512; e += 128) {
    int co = e >> 9, kk = e & 511;
    int kp = kk >> 6, ci = kk & 63;
    s_w[e] = cw[(co0 + co) * 512 + ci * 8 + kp];
  }
  const int w = tid >> 5, lane = tid & 31;
  const int lg = lane >> 4, ml = lane & 15;
  const float* __restrict__ base = p2 + (size_t)(b * 64) * PSTRIDE;
  float cmax = -3.4e38f;
  for (int blk = 0; blk < 16; ++blk) {
    const int Lb = blk * 64;
    __syncthreads();                // protect s_in from previous iteration
    for (int e = tid; e < 64 * 72; e += 128) {
      int ci = e / 72, t = e - ci * 72;     // consecutive threads -> consecutive t
      int pos = Lb - 3 + t;
      int pc = iclamp(pos, 0, L2 - 1);
      float v = base[ci * PSTRIDE + pc];
      s_in[t * 64 + ci] = (pos >= 0 && pos < L2) ? v : 0.f;
    }
    __syncthreads();
    const int lt = blk * 4 + w;
    if (lt < 63) {
      const int l0 = lt * 16;
      v8f acc = {};
#pragma unroll
      for (int kp = 0; kp < 8; ++kp) {
        const float* __restrict__ ir = s_in + (w * 16 + ml + kp) * 64 + 2 * lg;
        const float* __restrict__ br = s_w + ml * 512 + kp * 64 + 2 * lg;
#pragma unroll 8
        for (int c = 0; c < 64; c += 4) {
          v2f a = ld2(&ir[c]);
          v2f bb = ld2(&br[c]);
          acc = wmma4(a, bb, acc);
        }
      }
#pragma unroll
      for (int r = 0; r < 8; ++r) {
        int l = l0 + r + 8 * lg;
        if (l < L3) cmax = fmaxf(cmax, acc[r]);
      }
    }
  }
  cmax = fmaxf(cmax, __shfl_xor(cmax, 16, 32));
  if (lane < 16) s_red[w * 16 + ml] = cmax;
  __syncthreads();
  if (tid < 16) {
    float m = fmaxf(fmaxf(s_red[tid], s_red[16 + tid]),
                    fmaxf(s_red[32 + tid], s_red[48 + tid]));
    int co = co0 + tid;
    z[b * 224 + 128 + co] = fmaxf(m + cb[co], 0.f);
  }
}

// ------------------------------------------------------------- final 256->1 FC
__global__ void k_fc3(const float* __restrict__ a, const float* __restrict__ wv,
                      const float* __restrict__ bv, float* __restrict__ out) {
  int i = blockIdx.x * blockDim.x + threadIdx.x;
  if (i >= N_GRAPHS) return;
  float s = 0.f;
  for (int k = 0; k < 256; ++k) s += a[i * 256 + k] * wv[k];
  out[i] = s + bv[0];
}

// =============================================================== orchestration
static inline int cdiv(int a, int b) { return (a + b - 1) / b; }

extern "C" void kernel_launch(void* const* d_in, const int* in_sizes, int n_in,
                              void* d_out, int out_size, void* d_ws, size_t ws_size,
                              hipStream_t stream) {
  const float* x   = (const float*)d_in[0];
  const int*   ei  = (const int*)d_in[1];
  const int*   bat = (const int*)d_in[2];
  const int*   ps  = (const int*)d_in[3];
  const float *W1 = (const float*)d_in[4],  *b1 = (const float*)d_in[5],
              *g1 = (const float*)d_in[6],  *be1 = (const float*)d_in[7];
  const float *W2 = (const float*)d_in[8],  *b2 = (const float*)d_in[9],
              *g2 = (const float*)d_in[10], *be2 = (const float*)d_in[11];
  const float *W3 = (const float*)d_in[12], *b3 = (const float*)d_in[13],
              *g3 = (const float*)d_in[14], *be3 = (const float*)d_in[15];
  const float* emb = (const float*)d_in[16];
  const float *cw1 = (const float*)d_in[17], *cb1 = (const float*)d_in[18];
  const float *cw2 = (const float*)d_in[19], *cb2 = (const float*)d_in[20];
  const float *cw3 = (const float*)d_in[21], *cb3 = (const float*)d_in[22];
  const float *lw1 = (const float*)d_in[23], *lb1 = (const float*)d_in[24],
              *lg1 = (const float*)d_in[25], *lbe1 = (const float*)d_in[26];
  const float *lw2 = (const float*)d_in[27], *lb2 = (const float*)d_in[28],
              *lg2 = (const float*)d_in[29], *lbe2 = (const float*)d_in[30];
  const float *lw3 = (const float*)d_in[31], *lb3 = (const float*)d_in[32];

  const int* src = ei;
  const int* dst = ei + N_EDGES;

  float* ws = (float*)d_ws;
  size_t o = 0;
  float* deg  = ws + o; o += N_NODES;                 // -> dinv in place
  float* H    = ws + o; o += (size_t)N_NODES * 128;   // GEMM outputs
  float* ACT  = ws + o; o += (size_t)N_NODES * 128;   // aggregated activations
  float* mean = ws + o; o += 512;
  float* rstd = ws + o; o += 512;
  float* z    = ws + o; o += (size_t)N_GRAPHS * 224;  // [drug(128) | prot(96)]
  float* cnt  = ws + o; o += N_GRAPHS;
  float* t1   = ws + o; o += (size_t)N_GRAPHS * 512;
  float* t2   = ws + o; o += (size_t)N_GRAPHS * 256;
  float* p1   = ws + o; o += (size_t)N_GRAPHS * 32 * PSTRIDE;
  float* p2   = ws + o; o += (size_t)N_GRAPHS * 64 * PSTRIDE;
  (void)o;

  const int TB = 256;

  // ---- degrees -> dinv
  k_fill<<<cdiv(N_NODES, TB), TB, 0, stream>>>(deg, 1.0f, N_NODES);
  k_deg_count<<<cdiv(N_EDGES, TB), TB, 0, stream>>>(dst, deg, N_EDGES);
  k_rsqrt_inplace<<<cdiv(N_NODES, TB), TB, 0, stream>>>(deg, N_NODES);

  // ---- GCN layer 1 (9 -> 64)
  k_gemm<<<dim3(cdiv(N_NODES, 64), 4), 128, 0, stream>>>(x, W1, nullptr, H, N_NODES, 9, 64);
  k_agg_init<<<cdiv(N_NODES * 64, TB), TB, 0, stream>>>(H, deg, b1, ACT, 6);
  k_agg_edge<<<cdiv(N_EDGES * 64, TB), TB, 0, stream>>>(H, deg, src, dst, ACT, 6);
  k_bn_stats<<<64, 256, 0, stream>>>(ACT, mean, rstd, N_NODES, 64);
  k_bn_relu<<<cdiv(N_NODES * 64, TB), TB, 0, stream>>>(ACT, g1, be1, mean, rstd, N_NODES * 64, 63);

  // ---- GCN layer 2 (64 -> 128)
  k_gemm<<<dim3(cdiv(N_NODES, 64), 8), 128, 0, stream>>>(ACT, W2, nullptr, H, N_NODES, 64, 128);
  k_agg_init<<<cdiv(N_NODES * 128, TB), TB, 0, stream>>>(H, deg, b2, ACT, 7);
  k_agg_edge<<<cdiv(N_EDGES * 128, TB), TB, 0, stream>>>(H, deg, src, dst, ACT, 7);
  k_bn_stats<<<128, 256, 0, stream>>>(ACT, mean, rstd, N_NODES, 128);
  k_bn_relu<<<cdiv(N_NODES * 128, TB), TB, 0, stream>>>(ACT, g2, be2, mean, rstd, N_NODES * 128, 127);

  // ---- GCN layer 3 (128 -> 128)
  k_gemm<<<dim3(cdiv(N_NODES, 64), 8), 128, 0, stream>>>(ACT, W3, nullptr, H, N_NODES, 128, 128);
  k_agg_init<<<cdiv(N_NODES * 128, TB), TB, 0, stream>>>(H, deg, b3, ACT, 7);
  k_agg_edge<<<cdiv(N_EDGES * 128, TB), TB, 0, stream>>>(H, deg, src, dst, ACT, 7);
  k_bn_stats<<<128, 256, 0, stream>>>(ACT, mean, rstd, N_NODES, 128);
  k_bn_relu<<<cdiv(N_NODES * 128, TB), TB, 0, stream>>>(ACT, g3, be3, mean, rstd, N_NODES * 128, 127);

  // ---- global mean pool -> z[:, 0:128]
  hipMemsetAsync(z, 0, (size_t)N_GRAPHS * 224 * sizeof(float), stream);
  hipMemsetAsync(cnt, 0, (size_t)N_GRAPHS * sizeof(float), stream);
  k_pool<<<cdiv(N_NODES * 128, TB), TB, 0, stream>>>(ACT, bat, z, cnt);
  k_pool_div<<<cdiv(N_GRAPHS * 128, TB), TB, 0, stream>>>(z, cnt);

  // ---- protein encoder -> z[:, 128:224]
  k_conv1<<<dim3(16, 2, N_GRAPHS), 128, 0, stream>>>(ps, emb, cw1, cb1, p1);
  k_conv2<<<dim3(16, 4, N_GRAPHS), 128, 0, stream>>>(p1, cw2, cb2, p2);
  k_conv3max<<<dim3(6, N_GRAPHS), 128, 0, stream>>>(p2, cw3, cb3, z);

  // ---- regressor
  k_gemm<<<dim3(cdiv(N_GRAPHS, 64), 32), 128, 0, stream>>>(z, lw1, lb1, t1, N_GRAPHS, 224, 512);
  k_bn_stats<<<512, 256, 0, stream>>>(t1, mean, rstd, N_GRAPHS, 512);
  k_bn_relu<<<cdiv(N_GRAPHS * 512, TB), TB, 0, stream>>>(t1, lg1, lbe1, mean, rstd, N_GRAPHS * 512, 511);

  k_gemm<<<dim3(cdiv(N_GRAPHS, 64), 16), 128, 0, stream>>>(t1, lw2, lb2, t2, N_GRAPHS, 512, 256);
  k_bn_stats<<<256, 256, 0, stream>>>(t2, mean, rstd, N_GRAPHS, 256);
  k_bn_relu<<<cdiv(N_GRAPHS * 256, TB), TB, 0, stream>>>(t2, lg2, lbe2, mean, rstd, N_GRAPHS * 256, 255);

  k_fc3<<<cdiv(N_GRAPHS, TB), TB, 0, stream>>>(t2, lw3, lb3, (float*)d_out);
}